// QuantizedCodebook_41549513621707
// MI455X (gfx1250) — compile-verified
//
#include <hip/hip_runtime.h>
#include <hip/hip_bf16.h>

typedef __attribute__((ext_vector_type(2))) float v2f;
typedef __attribute__((ext_vector_type(8))) float v8f;

#define NUM_E 8192
#define DIM   256
#define NROWS 32768
#define BETA  0.25f
#define MBLK  128   // rows of X per workgroup (8 waves x 16 rows)
#define NTILE 32    // codes staged in LDS per stage (2 WMMA N-tiles)
#define NSTAGES (NUM_E / NTILE)
#define STAGE_BYTES (NTILE * DIM * 4)   // 32 KB

// ---------------------------------------------------------------------------
// Kernel 1: wnorm[k] = sum_d w[k][d]^2 ; also zero the histogram counts.
// ---------------------------------------------------------------------------
__global__ __launch_bounds__(256) void vq_wnorm_kernel(
    const float* __restrict__ W, float* __restrict__ wnorm,
    unsigned int* __restrict__ counts) {
  int i = blockIdx.x * 256 + threadIdx.x;
  if (i < NUM_E) {
    const float4* r = (const float4*)(W + (size_t)i * DIM);
    float s = 0.f;
#pragma unroll 8
    for (int j = 0; j < DIM / 4; ++j) {
      float4 v = r[j];
      s += v.x * v.x + v.y * v.y + v.z * v.z + v.w * v.w;
    }
    wnorm[i] = s;
    counts[i] = 0u;
  }
}

// ---------------------------------------------------------------------------
// Kernel 2: fused X@W^T (f32 WMMA 16x16x4) + running per-row argmin.
// A operand (16 rows x 256 dims per wave) lives in 128 VGPRs.  W tiles are
// staged into a triple-buffered LDS ring by GLOBAL_LOAD_ASYNC_TO_LDS_B128
// (ASYNCcnt-tracked), one barrier per 32-code stage.
// Score s = ||w||^2 - 2 x.w  (||x||^2 is constant per row -> irrelevant).
// ---------------------------------------------------------------------------
__global__ __launch_bounds__(256) void vq_argmin_kernel(
    const float* __restrict__ X, const float* __restrict__ W,
    const float* __restrict__ wnorm, int* __restrict__ idx_out) {
  __shared__ float wtile[3 * NTILE * DIM];  // 3 x 32 KB async ring

  const int tid  = threadIdx.x;
  const int wave = tid >> 5;
  const int lane = tid & 31;
  const int half = lane >> 4;   // selects K pair {0,1} vs {2,3} in A/B
  const int l16  = lane & 15;
  const int row0 = blockIdx.x * MBLK;

  // A matrix for this wave: rows row0+wave*16 .. +15, fully register-resident.
  // Lane layout for V_WMMA_F32_16X16X4_F32 A (16x4): M = l16, VGPR0/1 = K,K+1
  // with K base 2*half; a[kk] covers the K-step k = 4*kk.
  v2f a[DIM / 4];
  {
    const float* xrow = X + (size_t)(row0 + wave * 16 + l16) * DIM + 2 * half;
#pragma unroll
    for (int kk = 0; kk < DIM / 4; ++kk)
      a[kk] = *(const v2f*)&xrow[4 * kk];  // 8B aligned -> global_load_b64
  }

  float minv[8];
  int   mini[8];
#pragma unroll
  for (int j = 0; j < 8; ++j) { minv[j] = 3.4e38f; mini[j] = 0; }

  // Per-thread base addresses for the async staging: thread t owns the 8
  // 16-byte chunks at (t + 256*i)*16 within each 32 KB stage.
  const unsigned int lds0 = (unsigned int)(size_t)(&wtile[0]) + (unsigned)tid * 16u;
  const unsigned long long g0 = (unsigned long long)(size_t)W + (unsigned long long)tid * 16u;

  // Issue one stage's 8 async b128 copies (512 B per wave-instruction).
  auto issue_stage = [&](int s, int bufi) {
    const unsigned int l = lds0 + (unsigned)bufi * (unsigned)STAGE_BYTES;
    const unsigned long long g = g0 + (unsigned long long)s * STAGE_BYTES;
    asm volatile("global_load_async_to_lds_b128 %0, %1, off offset:0x0000" :: "v"(l), "v"(g) : "memory");
    asm volatile("global_load_async_to_lds_b128 %0, %1, off offset:0x1000" :: "v"(l), "v"(g) : "memory");
    asm volatile("global_load_async_to_lds_b128 %0, %1, off offset:0x2000" :: "v"(l), "v"(g) : "memory");
    asm volatile("global_load_async_to_lds_b128 %0, %1, off offset:0x3000" :: "v"(l), "v"(g) : "memory");
    asm volatile("global_load_async_to_lds_b128 %0, %1, off offset:0x4000" :: "v"(l), "v"(g) : "memory");
    asm volatile("global_load_async_to_lds_b128 %0, %1, off offset:0x5000" :: "v"(l), "v"(g) : "memory");
    asm volatile("global_load_async_to_lds_b128 %0, %1, off offset:0x6000" :: "v"(l), "v"(g) : "memory");
    asm volatile("global_load_async_to_lds_b128 %0, %1, off offset:0x7000" :: "v"(l), "v"(g) : "memory");
  };

  // Prefill two stages of the ring.
  issue_stage(0, 0);
  issue_stage(1, 1);

  const int b0 = l16 * DIM + 2 * half;          // codes n0 + 0..15
  const int b1 = (16 + l16) * DIM + 2 * half;   // codes n0 + 16..31

  int bi = 0;  // ring slot holding stage s
  for (int s = 0; s < NSTAGES; ++s) {
    const int n0 = s * NTILE;

    // Async loads complete in order: <=8 outstanding => stage s is in LDS.
    if (s + 1 < NSTAGES) {
      asm volatile("s_wait_asynccnt 0x8" ::: "memory");
    } else {
      asm volatile("s_wait_asynccnt 0x0" ::: "memory");
    }
    __syncthreads();  // all waves' stage-s data visible

    // Refill the slot last read at stage s-1 (all waves passed barrier(s)).
    if (s + 2 < NSTAGES) {
      int bi2 = bi + 2; if (bi2 >= 3) bi2 -= 3;
      issue_stage(s + 2, bi2);
    }
    if (s + 3 < NSTAGES)
      __builtin_prefetch(W + (size_t)(n0 + 3 * NTILE) * DIM + tid * 16, 0, 1);

    // 2 N-tiles x 64 K-steps = 128 WMMAs; A from regs, B from LDS.
    const float* buf = wtile + bi * (NTILE * DIM);
    v8f c0 = {}, c1 = {};
#pragma unroll
    for (int kk = 0; kk < DIM / 4; ++kk) {
      v2f vb0 = *(const v2f*)&buf[b0 + 4 * kk];
      v2f vb1 = *(const v2f*)&buf[b1 + 4 * kk];
      c0 = __builtin_amdgcn_wmma_f32_16x16x4_f32(false, a[kk], false, vb0,
                                                 (short)0, c0, false, false);
      c1 = __builtin_amdgcn_wmma_f32_16x16x4_f32(false, a[kk], false, vb1,
                                                 (short)0, c1, false, false);
    }

    // Running argmin in C layout: VGPR j <-> (row j+8*half, col l16).
    // Strict < with ascending code order == first-occurrence argmin.
    const float wn0 = wnorm[n0 + l16];
    const float wn1 = wnorm[n0 + 16 + l16];
    const int   cd0 = n0 + l16;
    const int   cd1 = n0 + 16 + l16;
#pragma unroll
    for (int j = 0; j < 8; ++j) {
      float s0 = wn0 - 2.0f * c0[j];
      if (s0 < minv[j]) { minv[j] = s0; mini[j] = cd0; }
      float s1 = wn1 - 2.0f * c1[j];
      if (s1 < minv[j]) { minv[j] = s1; mini[j] = cd1; }
    }

    bi = (bi == 2) ? 0 : bi + 1;
  }

  // Final per-row reduction across the 16 lanes holding that row's columns.
#pragma unroll
  for (int j = 0; j < 8; ++j) {
    float v = minv[j];
    int   i = mini[j];
#pragma unroll
    for (int m = 8; m >= 1; m >>= 1) {
      float ov = __shfl_xor(v, m, 16);
      int   oi = __shfl_xor(i, m, 16);
      if (ov < v || (ov == v && oi < i)) { v = ov; i = oi; }
    }
    if (l16 == 0) {
      int row = row0 + wave * 16 + j + 8 * half;
      idx_out[row] = i;
    }
  }
}

// ---------------------------------------------------------------------------
// Kernel 3: zero the 1 GiB one-hot encodings region (float2: start is only
// 8-byte aligned within d_out).
// ---------------------------------------------------------------------------
__global__ __launch_bounds__(256) void vq_zero_enc_kernel(float2* __restrict__ enc2,
                                                          long long n2) {
  long long i = (long long)blockIdx.x * 256 + threadIdx.x;
  const long long stride = (long long)gridDim.x * 256;
  float2 z; z.x = 0.f; z.y = 0.f;
  for (; i < n2; i += stride) enc2[i] = z;
}

// ---------------------------------------------------------------------------
// Kernel 4: gather quantized = w[idx], per-row squared-error partial,
// integer histogram.  One block per row, one thread per dim.
// ---------------------------------------------------------------------------
__global__ __launch_bounds__(256) void vq_gather_kernel(
    const float* __restrict__ X, const float* __restrict__ W,
    const int* __restrict__ idx, float* __restrict__ q_out,
    float* __restrict__ partial, unsigned int* __restrict__ counts) {
  const int row = blockIdx.x;
  const int t = threadIdx.x;
  const int k = idx[row];
  float xv = X[(size_t)row * DIM + t];
  float wv = W[(size_t)k * DIM + t];
  q_out[(size_t)row * DIM + t] = wv;
  float d = xv - wv;
  d *= d;
#pragma unroll
  for (int m = 16; m >= 1; m >>= 1) d += __shfl_xor(d, m, 32);
  __shared__ float s[8];
  if ((t & 31) == 0) s[t >> 5] = d;
  __syncthreads();
  if (t == 0) {
    float tot = 0.f;
#pragma unroll
    for (int i = 0; i < 8; ++i) tot += s[i];
    partial[row] = tot;
    atomicAdd(&counts[k], 1u);
  }
}

// ---------------------------------------------------------------------------
// Kernel 5: scatter the ones into the zeroed one-hot matrix.
// ---------------------------------------------------------------------------
__global__ __launch_bounds__(256) void vq_scatter_kernel(const int* __restrict__ idx,
                                                         float* __restrict__ enc) {
  int row = blockIdx.x * 256 + threadIdx.x;
  if (row < NROWS) enc[(size_t)row * NUM_E + idx[row]] = 1.0f;
}

// ---------------------------------------------------------------------------
// Kernel 6: finalize loss = (1+beta)*MSE and perplexity (deterministic sums).
// ---------------------------------------------------------------------------
__global__ __launch_bounds__(256) void vq_finalize_kernel(
    const float* __restrict__ partial, const unsigned int* __restrict__ counts,
    float* __restrict__ loss_out, float* __restrict__ perp_out) {
  const int t = threadIdx.x;
  float ls = 0.f;
  for (int i = t; i < NROWS; i += 256) ls += partial[i];
  float es = 0.f;
  for (int kk = t; kk < NUM_E; kk += 256) {
    float p = (float)counts[kk] * (1.0f / (float)NROWS);
    es += p * logf(p + 1e-10f);
  }
#pragma unroll
  for (int m = 16; m >= 1; m >>= 1) {
    ls += __shfl_xor(ls, m, 32);
    es += __shfl_xor(es, m, 32);
  }
  __shared__ float sl[8], se[8];
  if ((t & 31) == 0) { sl[t >> 5] = ls; se[t >> 5] = es; }
  __syncthreads();
  if (t == 0) {
    float L = 0.f, E = 0.f;
#pragma unroll
    for (int i = 0; i < 8; ++i) { L += sl[i]; E += se[i]; }
    *loss_out = (1.0f + BETA) * L / (float)(NROWS * DIM);
    *perp_out = expf(-E);
  }
}

// ---------------------------------------------------------------------------
// d_out layout (floats): [0]=loss, [1 .. 1+N*D)=quantized,
// [1+N*D]=perplexity, [2+N*D ..)=encodings (N*K).
// d_ws layout: int idx[N] | uint counts[K] | float wnorm[K] | float partial[N]
// ---------------------------------------------------------------------------
extern "C" void kernel_launch(void* const* d_in, const int* in_sizes, int n_in,
                              void* d_out, int out_size, void* d_ws, size_t ws_size,
                              hipStream_t stream) {
  const float* X = (const float*)d_in[0];
  const float* W = (const float*)d_in[1];
  float* out = (float*)d_out;

  char* ws = (char*)d_ws;
  int*          idx     = (int*)ws;                               // 128 KB
  unsigned int* counts  = (unsigned int*)(ws + NROWS * 4);        // 32 KB
  float*        wnorm   = (float*)(ws + NROWS * 4 + NUM_E * 4);   // 32 KB
  float*        partial = (float*)(ws + NROWS * 4 + 2 * NUM_E * 4); // 128 KB

  float* loss_out = out;
  float* q_out    = out + 1;
  float* perp_out = out + 1 + (size_t)NROWS * DIM;
  float* enc      = out + 2 + (size_t)NROWS * DIM;

  vq_wnorm_kernel<<<NUM_E / 256, 256, 0, stream>>>(W, wnorm, counts);
  vq_argmin_kernel<<<NROWS / MBLK, 256, 0, stream>>>(X, W, wnorm, idx);
  const long long n2 = (long long)NROWS * NUM_E / 2;
  vq_zero_enc_kernel<<<8192, 256, 0, stream>>>((float2*)enc, n2);
  vq_gather_kernel<<<NROWS, 256, 0, stream>>>(X, W, idx, q_out, partial, counts);
  vq_scatter_kernel<<<NROWS / 256, 256, 0, stream>>>(idx, enc);
  vq_finalize_kernel<<<1, 256, 0, stream>>>(partial, counts, loss_out, perp_out);
}